// GATLayer_59742995088048
// MI455X (gfx1250) — compile-verified
//
#include <hip/hip_runtime.h>
#include <cstdint>
#include <cstddef>

typedef __attribute__((ext_vector_type(2))) float v2f;
typedef __attribute__((ext_vector_type(8))) float v8f;

// -------- ordered-uint encoding for float atomic max --------
__device__ __forceinline__ unsigned f2ord(float f) {
  unsigned u = __float_as_uint(f);
  return (u & 0x80000000u) ? ~u : (u | 0x80000000u);
}
__device__ __forceinline__ float ord2f(unsigned u) {
  unsigned b = (u & 0x80000000u) ? (u ^ 0x80000000u) : ~u;
  return __uint_as_float(b);
}

// -------- kernel 0: zero accumulators, init global max --------
__global__ void gat_init_kernel(float* __restrict__ out, float* __restrict__ denom,
                                unsigned* __restrict__ gmax, int n_out, int n_nodes) {
  int i = blockIdx.x * blockDim.x + threadIdx.x;
  if (i < n_out) out[i] = 0.0f;
  if (i < n_nodes) denom[i] = 0.0f;
  if (i == 0) *gmax = f2ord(-INFINITY);
}

// -------- kernel 1: Wh = x @ W  (IN=128, OUT=64) via fp32 WMMA 16x16x4 --------
// One wave computes a 16x64 output tile (4 WMMA accumulators across N).
// W staged in LDS K-pair-interleaved: Wl2[p][n] = {W[2p][n], W[2p+1][n]}
// so each B fragment is one aligned 8-byte ds_load into a VGPR pair.
__global__ __launch_bounds__(256) void gat_gemm_kernel(const float* __restrict__ x,
                                                       const float* __restrict__ W,
                                                       float* __restrict__ Wh, int N) {
  __shared__ v2f Wl2[64 * 64];   // [k-pair p][n] -> (W[2p][n], W[2p+1][n]); 32 KB
  {
    float* Wf = (float*)Wl2;
    for (int d = threadIdx.x; d < 128 * 64; d += 256) {
      const int p  = d >> 7;          // k-pair index
      const int n  = (d >> 1) & 63;   // output column
      const int rr = d & 1;           // which k of the pair
      Wf[d] = W[(2 * p + rr) * 64 + n];
    }
  }
  __syncthreads();

  const int wave = threadIdx.x >> 5;
  const int lane = threadIdx.x & 31;
  const int r0 = blockIdx.x * 128 + wave * 16;
  if (r0 >= N) return;  // wave-uniform exit

  const int hi = lane >> 4;      // K-pair selector within a K=4 step
  const int lo = lane & 15;      // row (A) / col (B)
  int m = r0 + lo;
  if (m > N - 1) m = N - 1;      // clamp A loads for tail tile (EXEC stays all-ones)
  const float* __restrict__ arow = x + (size_t)m * 128;

  v8f acc0 = {}, acc1 = {}, acc2 = {}, acc3 = {};
  for (int kc = 0; kc < 128; kc += 4) {
    const int k0 = kc + 2 * hi;          // always even
    const int p  = (kc >> 1) + hi;       // LDS k-pair row
    v2f a;
    a.x = arow[k0];
    a.y = arow[k0 + 1];
    const v2f* __restrict__ wrow = &Wl2[p * 64];
    v2f b0 = wrow[lo +  0];
    v2f b1 = wrow[lo + 16];
    v2f b2 = wrow[lo + 32];
    v2f b3 = wrow[lo + 48];
    acc0 = __builtin_amdgcn_wmma_f32_16x16x4_f32(false, a, false, b0, (short)0, acc0, false, false);
    acc1 = __builtin_amdgcn_wmma_f32_16x16x4_f32(false, a, false, b1, (short)0, acc1, false, false);
    acc2 = __builtin_amdgcn_wmma_f32_16x16x4_f32(false, a, false, b2, (short)0, acc2, false, false);
    acc3 = __builtin_amdgcn_wmma_f32_16x16x4_f32(false, a, false, b3, (short)0, acc3, false, false);
  }

  // C/D layout: lane l, vgpr r -> row = 8*(l>>4) + r, col = (l&15) + 16*tile
  #pragma unroll
  for (int r = 0; r < 8; ++r) {
    const int row = r0 + hi * 8 + r;
    if (row < N) {
      float* __restrict__ o = Wh + (size_t)row * 64;
      o[lo +  0] = acc0[r];
      o[lo + 16] = acc1[r];
      o[lo + 32] = acc2[r];
      o[lo + 48] = acc3[r];
    }
  }
}

// -------- kernel 2: per-node scores s_src = Wh.a[:64], s_dst = Wh.a[64:] --------
// One wave32 per node; lane covers channels {l, l+32}; butterfly reduce.
__global__ __launch_bounds__(256) void gat_score_kernel(const float* __restrict__ Wh,
                                                        const float* __restrict__ a,
                                                        float* __restrict__ s_src,
                                                        float* __restrict__ s_dst, int N) {
  const int node = blockIdx.x * 8 + (threadIdx.x >> 5);
  const int lane = threadIdx.x & 31;
  if (node >= N) return;
  const float* __restrict__ row = Wh + (size_t)node * 64;
  const float v0 = row[lane], v1 = row[lane + 32];
  float ss = v0 * a[lane]      + v1 * a[lane + 32];
  float sd = v0 * a[lane + 64] + v1 * a[lane + 96];
  #pragma unroll
  for (int off = 16; off > 0; off >>= 1) {
    ss += __shfl_xor(ss, off, 32);
    sd += __shfl_xor(sd, off, 32);
  }
  if (lane == 0) { s_src[node] = ss; s_dst[node] = sd; }
}

// -------- kernel 3: e = leaky_relu(s_src[src]+s_dst[dst]); global max --------
__global__ __launch_bounds__(256) void gat_edge_max_kernel(const int* __restrict__ src,
                                                           const int* __restrict__ dst,
                                                           const float* __restrict__ s_src,
                                                           const float* __restrict__ s_dst,
                                                           float* __restrict__ e_buf,
                                                           unsigned* __restrict__ gmax, int E) {
  const int i = blockIdx.x * blockDim.x + threadIdx.x;
  float e = -INFINITY;
  if (i < E) {
    const float v = s_src[src[i]] + s_dst[dst[i]];
    e = (v > 0.0f) ? v : 0.2f * v;
    e_buf[i] = e;
  }
  __shared__ float red[256];
  red[threadIdx.x] = e;
  __syncthreads();
  #pragma unroll
  for (int s = 128; s > 0; s >>= 1) {
    if (threadIdx.x < (unsigned)s) red[threadIdx.x] = fmaxf(red[threadIdx.x], red[threadIdx.x + s]);
    __syncthreads();
  }
  if (threadIdx.x == 0) atomicMax(gmax, f2ord(red[0]));
}

// -------- kernel 4: e_exp = exp(e - gmax); denom[dst] += e_exp --------
__global__ __launch_bounds__(256) void gat_edge_exp_kernel(const int* __restrict__ dst,
                                                           float* __restrict__ e_buf,
                                                           float* __restrict__ denom,
                                                           const unsigned* __restrict__ gmax, int E) {
  const int i = blockIdx.x * blockDim.x + threadIdx.x;
  if (i >= E) return;
  const float gm = ord2f(*gmax);
  const float ex = __expf(e_buf[i] - gm);
  e_buf[i] = ex;
  atomicAdd(&denom[dst[i]], ex);
}

// -------- kernel 5: out[dst] += (e_exp / max(denom[dst],1e-10)) * Wh[src] --------
// One wave32 per edge; lane covers channels {l, l+32}.
__global__ __launch_bounds__(256) void gat_scatter_kernel(const int* __restrict__ src,
                                                          const int* __restrict__ dst,
                                                          const float* __restrict__ e_buf,
                                                          const float* __restrict__ denom,
                                                          const float* __restrict__ Wh,
                                                          float* __restrict__ out, int E) {
  const int eid = blockIdx.x * 8 + (threadIdx.x >> 5);
  const int lane = threadIdx.x & 31;
  if (eid >= E) return;
  const int s = src[eid];
  const int d = dst[eid];
  const float alpha = e_buf[eid] / fmaxf(denom[d], 1e-10f);
  const float* __restrict__ wr = Wh + (size_t)s * 64;
  float* __restrict__ o = out + (size_t)d * 64;
  atomicAdd(&o[lane],      alpha * wr[lane]);
  atomicAdd(&o[lane + 32], alpha * wr[lane + 32]);
}

extern "C" void kernel_launch(void* const* d_in, const int* in_sizes, int n_in,
                              void* d_out, int out_size, void* d_ws, size_t ws_size,
                              hipStream_t stream) {
  const float* x          = (const float*)d_in[0];
  const int*   edge_index = (const int*)d_in[1];
  const float* W          = (const float*)d_in[2];
  const float* a          = (const float*)d_in[3];

  const int OUT = in_sizes[3] / 2;       // 64
  const int IN  = in_sizes[2] / OUT;     // 128
  const int N   = in_sizes[0] / IN;      // 100000
  const int E   = in_sizes[1] / 2;       // 1600000
  const int* src = edge_index;
  const int* dst = edge_index + E;
  float* out = (float*)d_out;

  // workspace layout (256B aligned slices)
  char* ws = (char*)d_ws;
  size_t off = 0;
  auto carve = [&](size_t bytes) -> char* {
    char* p = ws + off;
    off = (off + bytes + 255) & ~(size_t)255;
    return p;
  };
  float*    Wh    = (float*)carve((size_t)N * OUT * sizeof(float));
  float*    s_src = (float*)carve((size_t)N * sizeof(float));
  float*    s_dst = (float*)carve((size_t)N * sizeof(float));
  float*    e_buf = (float*)carve((size_t)E * sizeof(float));
  float*    denom = (float*)carve((size_t)N * sizeof(float));
  unsigned* gmax  = (unsigned*)carve(sizeof(unsigned));
  (void)ws_size; (void)n_in;

  const int n_out = out_size;            // N * OUT
  const int init_elems = (n_out > N) ? n_out : N;

  gat_init_kernel<<<(init_elems + 255) / 256, 256, 0, stream>>>(out, denom, gmax, n_out, N);
  gat_gemm_kernel<<<(N + 127) / 128, 256, 0, stream>>>(x, W, Wh, N);
  gat_score_kernel<<<(N + 7) / 8, 256, 0, stream>>>(Wh, a, s_src, s_dst, N);
  gat_edge_max_kernel<<<(E + 255) / 256, 256, 0, stream>>>(src, dst, s_src, s_dst, e_buf, gmax, E);
  gat_edge_exp_kernel<<<(E + 255) / 256, 256, 0, stream>>>(dst, e_buf, denom, gmax, E);
  gat_scatter_kernel<<<(E + 7) / 8, 256, 0, stream>>>(src, dst, e_buf, denom, Wh, out, E);
}